// EcologicalHeteroGNN_46222438039935
// MI455X (gfx1250) — compile-verified
//
#include <hip/hip_runtime.h>
#include <math.h>

#define N_SP 10000
#define N_LOC 50000
#define E_OBS 500000
#define E_NEAR 400000
#define NUM_GROUPS 16
#define EMB 48
#define LOC_IN 64
#define HID 128
#define HEADS 4
#define CH 32
#define NEG_SLOPE 0.2f

typedef __attribute__((ext_vector_type(2))) float v2f;
typedef __attribute__((ext_vector_type(8))) float v8f;

static inline int cdiv(long long a, long long b) { return (int)((a + b - 1) / b); }

// ---------------- feature build ----------------
__global__ void k_sp_feat(const int* __restrict__ idx, const float* __restrict__ emb,
                          const float* __restrict__ grp, float* __restrict__ out) {
  int t = blockIdx.x * blockDim.x + threadIdx.x;
  if (t >= N_SP * 64) return;
  int i = t >> 6, c = t & 63;
  out[t] = (c < EMB) ? emb[idx[i] * EMB + c] : grp[i * NUM_GROUPS + (c - EMB)];
}

// ---------------- f32 WMMA GEMM: C[N,128] = X[N,K] @ W[K,128] (+bias) ----------------
// block = 256 threads = 8 waves; wave w owns column tile w (16 cols) of a 32-row tile.
// grid.x = ceil(N/32). A rows are CLAMPED (not masked) so the inner loop is branch-free;
// clamped rows produce garbage accumulators that are never stored.
__global__ void k_gemm_wmma(const float* __restrict__ X, const float* __restrict__ W,
                            const float* __restrict__ bias, float* __restrict__ C,
                            int N, int K) {
  const int M = HID;
  int lane = threadIdx.x & 31;
  int col0 = (threadIdx.x >> 5) * 16;
  int row0 = blockIdx.x * 32;
  int half = lane >> 4;    // lanes 16-31 hold K=2,3 of each 4-step
  int l16  = lane & 15;
  int n = col0 + l16;
  int m0 = row0 + l16;
  int m1 = m0 + 16;
  int m0c = (m0 < N - 1) ? m0 : (N - 1);
  int m1c = (m1 < N - 1) ? m1 : (N - 1);
  const float* x0 = X + (size_t)m0c * K;
  const float* x1 = X + (size_t)m1c * K;
  v8f acc0 = {};
  v8f acc1 = {};
  for (int k = 0; k < K; k += 4) {
    int ka = k + half * 2;
    v2f b;
    b.x = W[(size_t)ka * M + n];
    b.y = W[(size_t)(ka + 1) * M + n];
    v2f a0 = *(const v2f*)(x0 + ka);   // 8B-aligned: ka even, K multiple of 4
    v2f a1 = *(const v2f*)(x1 + ka);
    acc0 = __builtin_amdgcn_wmma_f32_16x16x4_f32(false, a0, false, b, (short)0, acc0,
                                                 false, false);
    acc1 = __builtin_amdgcn_wmma_f32_16x16x4_f32(false, a1, false, b, (short)0, acc1,
                                                 false, false);
  }
  float bv = bias ? bias[n] : 0.f;
  int mb = row0 + half * 8;   // C layout: VGPR r -> tile row (half*8 + r), col = l16
#pragma unroll
  for (int r = 0; r < 8; ++r) {
    int mm0 = mb + r;
    if (mm0 < N) C[(size_t)mm0 * M + n] = acc0[r] + bv;
    int mm1 = mm0 + 16;
    if (mm1 < N) C[(size_t)mm1 * M + n] = acc1[r] + bv;
  }
}

// ---------------- attention scalars ----------------
__global__ void k_alpha(const float* __restrict__ xP, const float* __restrict__ avec,
                        float* __restrict__ al, int N) {
  int t = blockIdx.x * blockDim.x + threadIdx.x;
  if (t >= N * HEADS) return;
  int nn = t >> 2, h = t & 3;
  const float* x = xP + (size_t)nn * HID + h * CH;
  const float* a = avec + h * CH;
  float s = 0.f;
#pragma unroll
  for (int c = 0; c < CH; ++c) s = fmaf(x[c], a[c], s);
  al[t] = s;
}

__global__ void k_fill(float* __restrict__ p, float v, int n) {
  int t = blockIdx.x * blockDim.x + threadIdx.x;
  if (t < n) p[t] = v;
}

__device__ __forceinline__ float leaky(float v) { return v > 0.f ? v : NEG_SLOPE * v; }

__device__ __forceinline__ void atomicMaxF(float* addr, float v) {
  if (v >= 0.f) atomicMax((int*)addr, __float_as_int(v));
  else          atomicMin((unsigned int*)addr, __float_as_uint(v));
}

// edges e >= nE are synthesized self-loops (src=dst=e-nE)
__global__ void k_edge_max(const int* __restrict__ src, const int* __restrict__ dst,
                           int nE, int nSelf,
                           const float* __restrict__ al_s, const float* __restrict__ al_d,
                           float* __restrict__ emax) {
  int t = blockIdx.x * blockDim.x + threadIdx.x;
  if (t >= (nE + nSelf) * HEADS) return;
  int e = t >> 2, h = t & 3;
  int s, d;
  if (e < nE) { s = src[e]; d = dst[e]; } else { s = e - nE; d = s; }
  float v = leaky(al_s[s * HEADS + h] + al_d[d * HEADS + h]);
  atomicMaxF(&emax[d * HEADS + h], v);
}

__global__ void k_edge_denom(const int* __restrict__ src, const int* __restrict__ dst,
                             int nE, int nSelf,
                             const float* __restrict__ al_s, const float* __restrict__ al_d,
                             const float* __restrict__ emax, float* __restrict__ denom) {
  int t = blockIdx.x * blockDim.x + threadIdx.x;
  if (t >= (nE + nSelf) * HEADS) return;
  int e = t >> 2, h = t & 3;
  int s, d;
  if (e < nE) { s = src[e]; d = dst[e]; } else { s = e - nE; d = s; }
  float v = leaky(al_s[s * HEADS + h] + al_d[d * HEADS + h]);
  float m = emax[d * HEADS + h];
  m = (m > -1e30f) ? m : 0.f;
  atomicAdd(&denom[d * HEADS + h], __expf(v - m));
}

// one thread per (edge, 4-channel group): one b128 gather + 4 f32 atomic adds
__global__ void k_edge_msg(const int* __restrict__ src, const int* __restrict__ dst,
                           int nE, int nSelf,
                           const float* __restrict__ al_s, const float* __restrict__ al_d,
                           const float* __restrict__ emax, const float* __restrict__ denom,
                           const float* __restrict__ xs, float* __restrict__ out) {
  long long t = (long long)blockIdx.x * blockDim.x + threadIdx.x;
  long long total = (long long)(nE + nSelf) * (HID / 4);
  if (t >= total) return;
  int e  = (int)(t >> 5);
  int ch = (int)(t & 31) * 4;
  int h  = ch >> 5;
  int s, d;
  if (e < nE) { s = src[e]; d = dst[e]; } else { s = e - nE; d = s; }
  float v = leaky(al_s[s * HEADS + h] + al_d[d * HEADS + h]);
  float m = emax[d * HEADS + h];
  m = (m > -1e30f) ? m : 0.f;
  float alpha = __expf(v - m) / (denom[d * HEADS + h] + 1e-16f);
  float4 xv = *(const float4*)(xs + (size_t)s * HID + ch);
  float* op = out + (size_t)d * HID + ch;
  atomicAdd(op + 0, alpha * xv.x);
  atomicAdd(op + 1, alpha * xv.y);
  atomicAdd(op + 2, alpha * xv.z);
  atomicAdd(op + 3, alpha * xv.w);
}

// out = [relu]( acc + b0 [+ b1] + hprev )
__global__ void k_finalize(const float* __restrict__ acc, const float* __restrict__ hprev,
                           const float* __restrict__ b0, const float* __restrict__ b1,
                           float* __restrict__ out, int N, int relu) {
  int t = blockIdx.x * blockDim.x + threadIdx.x;
  if (t >= N * HID) return;
  int col = t & 127;
  float v = acc[t] + hprev[t] + b0[col];
  if (b1) v += b1[col];
  if (relu) v = fmaxf(v, 0.f);
  out[t] = v;
}

// ---------------- host orchestration ----------------
static void launch_gemm(const float* X, const float* W, const float* b, float* C,
                        int N, int K, hipStream_t s) {
  k_gemm_wmma<<<dim3(cdiv(N, 32)), 256, 0, s>>>(X, W, b, C, N, K);
}

static void run_conv(const int* srcArr, const int* dstArr, int nE, int nSelf, int nDst,
                     const float* al_s, const float* al_d,
                     float* emax, float* denom,
                     const float* xs, float* out_acc, int clear_out,
                     hipStream_t s) {
  k_fill<<<cdiv(nDst * HEADS, 256), 256, 0, s>>>(emax, -INFINITY, nDst * HEADS);
  k_fill<<<cdiv(nDst * HEADS, 256), 256, 0, s>>>(denom, 0.f, nDst * HEADS);
  if (clear_out)
    k_fill<<<cdiv(nDst * HID, 256), 256, 0, s>>>(out_acc, 0.f, nDst * HID);
  int nTot = nE + nSelf;
  k_edge_max<<<cdiv((long long)nTot * HEADS, 256), 256, 0, s>>>(
      srcArr, dstArr, nE, nSelf, al_s, al_d, emax);
  k_edge_denom<<<cdiv((long long)nTot * HEADS, 256), 256, 0, s>>>(
      srcArr, dstArr, nE, nSelf, al_s, al_d, emax, denom);
  k_edge_msg<<<cdiv((long long)nTot * (HID / 4), 256), 256, 0, s>>>(
      srcArr, dstArr, nE, nSelf, al_s, al_d, emax, denom, xs, out_acc);
}

extern "C" void kernel_launch(void* const* d_in, const int* in_sizes, int n_in,
                              void* d_out, int out_size, void* d_ws, size_t ws_size,
                              hipStream_t stream) {
  const int*   species_idx = (const int*)d_in[0];
  const float* group_feat  = (const float*)d_in[1];
  const float* loc_x       = (const float*)d_in[2];
  const int*   ei_obs      = (const int*)d_in[3];   // [2, E_OBS]: src then dst
  const int*   ei_obsat    = (const int*)d_in[4];
  const int*   ei_near     = (const int*)d_in[5];
  const float* emb_table   = (const float*)d_in[6];
  const float* sp_w        = (const float*)d_in[7];
  const float* sp_b        = (const float*)d_in[8];
  const float* loc_w       = (const float*)d_in[9];
  const float* loc_b       = (const float*)d_in[10];
  const float* gat_w       = (const float*)d_in[11]; // [2,3,128,128]
  const float* gat_as      = (const float*)d_in[12]; // [2,3,4,32]
  const float* gat_ad      = (const float*)d_in[13];
  const float* gat_b       = (const float*)d_in[14]; // [2,3,128]

  float* ws = (float*)d_ws;
  size_t off = 0;
  float* sp_feat = ws + off; off += (size_t)N_SP * 64;
  float* h_sp    = ws + off; off += (size_t)N_SP * HID;
  float* new_sp  = ws + off; off += (size_t)N_SP * HID;
  float* xP_sp   = ws + off; off += (size_t)N_SP * HID;
  float* h_loc   = ws + off; off += (size_t)N_LOC * HID;
  float* new_loc = ws + off; off += (size_t)N_LOC * HID;
  float* xP_loc  = ws + off; off += (size_t)N_LOC * HID;
  float* al_s    = ws + off; off += (size_t)N_LOC * HEADS;
  float* al_d    = ws + off; off += (size_t)N_LOC * HEADS;
  float* emax    = ws + off; off += (size_t)N_LOC * HEADS;
  float* denom   = ws + off; off += (size_t)N_LOC * HEADS;

  // input projections
  k_sp_feat<<<cdiv(N_SP * 64, 256), 256, 0, stream>>>(species_idx, emb_table,
                                                      group_feat, sp_feat);
  launch_gemm(sp_feat, sp_w, sp_b, h_sp, N_SP, LOC_IN, stream);   // K = 48+16 = 64
  launch_gemm(loc_x, loc_w, loc_b, h_loc, N_LOC, LOC_IN, stream);

  float* out_sp_final  = (float*)d_out;
  float* out_loc_final = (float*)d_out + (size_t)N_SP * HID;

  for (int L = 0; L < 2; ++L) {
    const float* W0 = gat_w + (size_t)(L * 3 + 0) * HID * HID;
    const float* W1 = W0 + (size_t)HID * HID;
    const float* W2 = W1 + (size_t)HID * HID;
    const float* as0 = gat_as + (size_t)(L * 3 + 0) * HEADS * CH;
    const float* as1 = as0 + HEADS * CH;
    const float* as2 = as1 + HEADS * CH;
    const float* ad0 = gat_ad + (size_t)(L * 3 + 0) * HEADS * CH;
    const float* ad1 = ad0 + HEADS * CH;
    const float* ad2 = ad1 + HEADS * CH;
    const float* b0 = gat_b + (size_t)(L * 3 + 0) * HID;
    const float* b1 = b0 + HID;
    const float* b2 = b1 + HID;

    // Conv A: species <- observes  (src = location, dst = species)
    launch_gemm(h_loc, W0, nullptr, xP_loc, N_LOC, HID, stream);
    launch_gemm(h_sp,  W0, nullptr, xP_sp,  N_SP,  HID, stream);
    k_alpha<<<cdiv(N_LOC * HEADS, 256), 256, 0, stream>>>(xP_loc, as0, al_s, N_LOC);
    k_alpha<<<cdiv(N_SP  * HEADS, 256), 256, 0, stream>>>(xP_sp,  ad0, al_d, N_SP);
    run_conv(ei_obs, ei_obs + E_OBS, E_OBS, 0, N_SP, al_s, al_d, emax, denom,
             xP_loc, new_sp, /*clear=*/1, stream);

    // Conv B: location <- observed_at  (src = species, dst = location)
    launch_gemm(h_sp,  W1, nullptr, xP_sp,  N_SP,  HID, stream);
    launch_gemm(h_loc, W1, nullptr, xP_loc, N_LOC, HID, stream);
    k_alpha<<<cdiv(N_SP  * HEADS, 256), 256, 0, stream>>>(xP_sp,  as1, al_s, N_SP);
    k_alpha<<<cdiv(N_LOC * HEADS, 256), 256, 0, stream>>>(xP_loc, ad1, al_d, N_LOC);
    run_conv(ei_obsat, ei_obsat + E_OBS, E_OBS, 0, N_LOC, al_s, al_d, emax, denom,
             xP_sp, new_loc, /*clear=*/1, stream);

    // Conv C: location <- nearby (+ self loops), src/dst share projection
    launch_gemm(h_loc, W2, nullptr, xP_loc, N_LOC, HID, stream);
    k_alpha<<<cdiv(N_LOC * HEADS, 256), 256, 0, stream>>>(xP_loc, as2, al_s, N_LOC);
    k_alpha<<<cdiv(N_LOC * HEADS, 256), 256, 0, stream>>>(xP_loc, ad2, al_d, N_LOC);
    run_conv(ei_near, ei_near + E_NEAR, E_NEAR, N_LOC, N_LOC, al_s, al_d, emax, denom,
             xP_loc, new_loc, /*clear=*/0, stream);

    // residual (+ReLU on layer 0); layer 1 writes the final outputs
    float* osp = (L == 0) ? h_sp  : out_sp_final;
    float* olc = (L == 0) ? h_loc : out_loc_final;
    k_finalize<<<cdiv(N_SP * HID, 256), 256, 0, stream>>>(
        new_sp, h_sp, b0, nullptr, osp, N_SP, L == 0);
    k_finalize<<<cdiv(N_LOC * HID, 256), 256, 0, stream>>>(
        new_loc, h_loc, b1, b2, olc, N_LOC, L == 0);
  }
}